// chamfer_loss_with_intensity_63127429316931
// MI455X (gfx1250) — compile-verified
//
#include <hip/hip_runtime.h>

// Chamfer + intensity loss for MI455X (gfx1250, wave32, WMMA).
//
// d2[i][j] = a2[i] + b2[j] - 2*dot(q_i, db_j), computed with WMMA F32 16x16x4:
//   A'[i] = (-2x,-2y,-2z, 1),  B'[j] = (bx,by,bz,b2_j)
//   => tile = A' x B' = -2*dot + b2[j];  a2[i] added post-argmin (row-constant).
//
// Both directions in ONE launch (block-range split, ~1250 live waves).
// db is staged per-block through LDS in 1024-point chunks (b2 precomputed once
// per chunk), with the next chunk loaded into registers during processing
// (double-buffer: global latency hidden behind ~64 tiles of WMMA work).
// Deterministic: per-wave partials + fixed-assignment wave-parallel final sum.

typedef __attribute__((ext_vector_type(2))) float v2f;
typedef __attribute__((ext_vector_type(8))) float v8f;

#define RB    2                 // row-blocks (16 rows each) per wave
#define WPB   8                 // waves per block (256 threads)
#define CHUNK 1024              // db points staged per block iteration
#define LPT   (CHUNK / 256)     // staged points per thread (= 4)

__global__ __launch_bounds__(256) void chamfer_both_kernel(
    const float* __restrict__ adv, int N,
    const float* __restrict__ ori, int M,
    int blocks0,
    float* __restrict__ part0,      // [2 * blocks0*WPB]
    float* __restrict__ part1)      // [2 * blocks1*WPB]
{
  __shared__ float ldsbuf[CHUNK * 4];   // (x, y, z, b2) per staged point, 16 KB

  // ---- pass selection by block range (whole block same pass -> uniform)
  const bool passB = (int)blockIdx.x >= blocks0;
  const float* __restrict__ q  = passB ? ori : adv;   // queries  [Nq,4]
  const float* __restrict__ db = passB ? adv : ori;   // database [Ndb,4]
  const int Nq  = passB ? M : N;
  const int Ndb = passB ? N : M;
  float* __restrict__ partial = passB ? part1 : part0;
  const int blockInPass = passB ? ((int)blockIdx.x - blocks0) : (int)blockIdx.x;

  const int tid     = threadIdx.x;
  const int lane    = tid & 31;
  const int wave    = blockInPass * WPB + (tid >> 5);
  const int col     = lane & 15;
  const bool lowHalf = lane < 16;
  const int rowbase = wave * (16 * RB);

  // ---- A fragments (query rows), f32 16x4 layout:
  //   VGPR0: lanes 0-15 -> K=0, lanes 16-31 -> K=2
  //   VGPR1: lanes 0-15 -> K=1, lanes 16-31 -> K=3
  v2f   afrag[RB];
  float qa2[RB], qint[RB];
#pragma unroll
  for (int rb = 0; rb < RB; ++rb) {
    int row = rowbase + rb * 16 + col;
    int rc  = row < Nq ? row : Nq - 1;   // clamp; guarded at contribution time
    const float4 p = *reinterpret_cast<const float4*>(q + 4 * rc);
    qa2[rb]  = p.x * p.x + p.y * p.y + p.z * p.z;
    qint[rb] = p.w;
    afrag[rb][0] = lowHalf ? (-2.0f * p.x) : (-2.0f * p.z);
    afrag[rb][1] = lowHalf ? (-2.0f * p.y) : 1.0f;
  }

  // Running min/argmin in the C-matrix layout:
  //   slot [rb][v] @ lane l covers row (rowbase + rb*16 + v + 8*(l>=16)),
  //   columns j with j%16 == l%16.
  float runmin[RB][8];
  int   runidx[RB][8];
#pragma unroll
  for (int rb = 0; rb < RB; ++rb)
#pragma unroll
    for (int v = 0; v < 8; ++v) { runmin[rb][v] = 3.0e38f; runidx[rb][v] = 0; }

  auto ldPoint = [&](int idx) -> float4 {
    int ic = idx < Ndb ? idx : Ndb - 1;  // clamp: duplicate last point (safe)
    return *reinterpret_cast<const float4*>(db + 4 * ic);
  };

  // ---- stage chunk 0 into registers
  float4 stage[LPT];
#pragma unroll
  for (int c = 0; c < LPT; ++c) stage[c] = ldPoint(tid + c * 256);

  const int nchunks = (Ndb + CHUNK - 1) / CHUNK;
  // per-lane LDS read base: point col, low half -> (x,y), high half -> (z,b2)
  const float* lanePtr = &ldsbuf[col * 4 + (lowHalf ? 0 : 2)];

  for (int k = 0; k < nchunks; ++k) {
    // ---- write staged chunk k to LDS as (x,y,z,b2); conflict-free b128 stores
#pragma unroll
    for (int c = 0; c < LPT; ++c) {
      float4 p = stage[c];
      float b2 = p.x * p.x + p.y * p.y + p.z * p.z;
      *reinterpret_cast<float4*>(&ldsbuf[(tid + c * 256) * 4]) =
          make_float4(p.x, p.y, p.z, b2);
    }
    __syncthreads();

    // ---- issue loads for chunk k+1 now; their wait lands at next ds_store,
    //      covered by the ~64-tile WMMA loop below (register double-buffer)
    const int nextbase = (k + 1 < nchunks ? k + 1 : k) * CHUNK;
#pragma unroll
    for (int c = 0; c < LPT; ++c) stage[c] = ldPoint(nextbase + tid + c * 256);

    // ---- process chunk k from LDS
    const int chunkstart = k * CHUNK;
    int rem   = Ndb - chunkstart;
    int tiles = rem < CHUNK ? rem : CHUNK;
    tiles = (tiles + 15) >> 4;

    for (int tl = 0; tl < tiles; ++tl) {
      // ds_load_b64: lanes 0-15 get (x,y)=K0,K1; lanes 16-31 get (z,b2)=K2,K3
      const float2 bf = *reinterpret_cast<const float2*>(lanePtr + tl * 64);
      v2f bfrag; bfrag[0] = bf.x; bfrag[1] = bf.y;
      int j = chunkstart + tl * 16 + col;
      j = j < Ndb ? j : Ndb - 1;

#pragma unroll
      for (int rb = 0; rb < RB; ++rb) {
        v8f acc = {0.f, 0.f, 0.f, 0.f, 0.f, 0.f, 0.f, 0.f};
        acc = __builtin_amdgcn_wmma_f32_16x16x4_f32(
            /*neg_a=*/false, afrag[rb], /*neg_b=*/false, bfrag,
            /*c_mod=*/(short)0, acc, /*reuse_a=*/false, /*reuse_b=*/false);
#pragma unroll
        for (int v = 0; v < 8; ++v) {
          float d2  = acc[v];
          bool  hit = d2 < runmin[rb][v];
          runmin[rb][v] = hit ? d2 : runmin[rb][v];
          runidx[rb][v] = hit ? j  : runidx[rb][v];
        }
      }
    }
    __syncthreads();   // LDS reuse barrier before next chunk's stores
  }

  // ---- per-row argmin: xor-shuffle within each 16-lane half (masks 8..1)
  float sumMin = 0.f, sumInt = 0.f;
#pragma unroll
  for (int rb = 0; rb < RB; ++rb) {
#pragma unroll
    for (int v = 0; v < 8; ++v) {
      float val = runmin[rb][v];
      int   idx = runidx[rb][v];
#pragma unroll
      for (int m = 8; m >= 1; m >>= 1) {
        float oval = __shfl_xor(val, m, 32);
        int   oidx = __shfl_xor(idx, m, 32);
        if (oval < val) { val = oval; idx = oidx; }
      }
      float dbint = db[4 * idx + 3];   // winner's intensity (idx valid, all lanes)
      // owner lanes: lane v    -> row rowbase+rb*16+v     (low-half result)
      //              lane v+24 -> row rowbase+rb*16+v+8   (high-half result)
      // these lanes also hold the matching query's qa2/qint.
      int  rowReal = rowbase + rb * 16 + (lowHalf ? v : v + 8);
      bool own     = (lane == v) || (lane == v + 24);
      if (own && rowReal < Nq) {
        sumMin += val + qa2[rb];
        float diff = qint[rb] - dbint;
        sumInt += diff * diff;
      }
    }
  }

  // ---- wave-level sum, write fixed per-wave slot (deterministic)
#pragma unroll
  for (int m = 16; m >= 1; m >>= 1) {
    sumMin += __shfl_xor(sumMin, m, 32);
    sumInt += __shfl_xor(sumInt, m, 32);
  }
  if (lane == 0) {
    partial[2 * wave]     = sumMin;
    partial[2 * wave + 1] = sumInt;
  }
}

// Wave-parallel, fixed-assignment (deterministic) final reduction.
__global__ __launch_bounds__(32) void chamfer_finalize(
    const float* __restrict__ p0, int w0,
    const float* __restrict__ p1, int w1,
    float invN, float invM,
    float* __restrict__ out)
{
  const int lane = threadIdx.x;   // blockDim == 32
  float sm0 = 0.f, si0 = 0.f, sm1 = 0.f, si1 = 0.f;
  for (int i = lane; i < w0; i += 32) { sm0 += p0[2 * i]; si0 += p0[2 * i + 1]; }
  for (int i = lane; i < w1; i += 32) { sm1 += p1[2 * i]; si1 += p1[2 * i + 1]; }
#pragma unroll
  for (int m = 16; m >= 1; m >>= 1) {
    sm0 += __shfl_xor(sm0, m, 32);
    si0 += __shfl_xor(si0, m, 32);
    sm1 += __shfl_xor(sm1, m, 32);
    si1 += __shfl_xor(si1, m, 32);
  }
  if (lane == 0) {
    float chamfer   = sm0 * invN + sm1 * invM;            // mean + mean
    float intensity = 0.5f * (si0 * invN + si1 * invM);   // (mean+mean)/2
    out[0] = chamfer * 1.0f /*LOSS_WEIGHT*/ + intensity * 0.5f /*INTENSITY_WEIGHT*/;
  }
}

extern "C" void kernel_launch(void* const* d_in, const int* in_sizes, int n_in,
                              void* d_out, int out_size, void* d_ws, size_t ws_size,
                              hipStream_t stream) {
  const float* adv = (const float*)d_in[0];   // [N,4] float32
  const float* ori = (const float*)d_in[1];   // [M,4] float32
  const int N = in_sizes[0] / 4;
  const int M = in_sizes[1] / 4;
  float* out = (float*)d_out;
  float* ws  = (float*)d_ws;

  const int rowsPerWave = 16 * RB;
  const int waves0  = (N + rowsPerWave - 1) / rowsPerWave;
  const int blocks0 = (waves0 + WPB - 1) / WPB;
  const int slots0  = blocks0 * WPB;
  const int waves1  = (M + rowsPerWave - 1) / rowsPerWave;
  const int blocks1 = (waves1 + WPB - 1) / WPB;
  const int slots1  = blocks1 * WPB;

  float* p0 = ws;                  // [2*slots0]
  float* p1 = ws + 2 * slots0;     // [2*slots1]   (~10 KB workspace total)

  chamfer_both_kernel<<<blocks0 + blocks1, 256, 0, stream>>>(
      adv, N, ori, M, blocks0, p0, p1);

  chamfer_finalize<<<1, 32, 0, stream>>>(p0, slots0, p1, slots1,
                                         1.0f / (float)N, 1.0f / (float)M, out);
}